// BCE_sigmoid_negtive_bias_all_48095043781157
// MI455X (gfx1250) — compile-verified
//
#include <hip/hip_runtime.h>
#include <hip/hip_bf16.h>

// ---------------- problem constants ----------------
#define NROWS 2097152
#define AU    12
#define NBINS 1024          // per-column histogram bins over rand_scores in [0,1)
#define CAP   4096          // per-column capacity for boundary-bin candidates
#define NPART 1024          // partial sums (one per final-pass block)

// workspace layout (bytes)
#define OFF_POS    0                         // int[12]
#define OFF_NEG    48                        // int[12]
#define OFF_KDROP  96                        // int[12]
#define OFF_PFLAG  144                       // int[12]
#define OFF_BBIN   192                       // int[12]
#define OFF_RESID  240                       // int[12]
#define OFF_TIECNT 288                       // int[12]
#define OFF_RROW   336                       // int[12]
#define OFF_RATIO  384                       // float[12]
#define OFF_TTHR   432                       // float[12]
#define OFF_HIST   512                       // int[12*NBINS]
#define OFF_TIES_S (OFF_HIST + AU*NBINS*4)   // float[12*CAP]
#define OFF_TIES_R (OFF_TIES_S + AU*CAP*4)   // int[12*CAP]
#define OFF_PART   (OFF_TIES_R + AU*CAP*4)   // float[NPART]
#define WS_ZERO    (OFF_HIST + AU*NBINS*4)   // zero counters+params+hist each call

typedef float v2f __attribute__((ext_vector_type(2)));
typedef float v8f __attribute__((ext_vector_type(8)));

__constant__ float c_balance[AU] = {0.2f,0.3f,0.2f,0.2f,0.5f,0.2f,0.5f,0.2f,0.1f,0.5f,0.2f,0.3f};

__device__ __forceinline__ void load_row_i(const int* p, int base, int (&l)[AU]) {
  const int4* q = (const int4*)(p + base);
  int4 a = q[0], b = q[1], c = q[2];
  l[0]=a.x; l[1]=a.y; l[2]=a.z; l[3]=a.w;
  l[4]=b.x; l[5]=b.y; l[6]=b.z; l[7]=b.w;
  l[8]=c.x; l[9]=c.y; l[10]=c.z; l[11]=c.w;
}
__device__ __forceinline__ void load_row_f(const float* p, int base, float (&s)[AU]) {
  const float4* q = (const float4*)(p + base);
  float4 a = q[0], b = q[1], c = q[2];
  s[0]=a.x; s[1]=a.y; s[2]=a.z; s[3]=a.w;
  s[4]=b.x; s[5]=b.y; s[6]=b.z; s[7]=b.w;
  s[8]=c.x; s[9]=c.y; s[10]=c.z; s[11]=c.w;
}
__device__ __forceinline__ int score_bin(float s) {
  int b = (int)(s * (float)NBINS);
  return b > (NBINS-1) ? (NBINS-1) : (b < 0 ? 0 : b);
}

// ---- pass 1: fused per-column pos/neg counts + histogram of negative scores ----
__global__ void k_count_hist(const int* __restrict__ labels, const float* __restrict__ scores,
                             int* __restrict__ g_pos, int* __restrict__ g_neg,
                             int* __restrict__ g_hist) {
  __shared__ int sh_hist[AU * NBINS];
  __shared__ int sh_cnt[2 * AU];
  for (int i = threadIdx.x; i < AU * NBINS; i += blockDim.x) sh_hist[i] = 0;
  if (threadIdx.x < 2 * AU) sh_cnt[threadIdx.x] = 0;
  __syncthreads();

  int pc[AU], nc[AU];
  #pragma unroll
  for (int c = 0; c < AU; c++) { pc[c] = 0; nc[c] = 0; }

  const int stride = gridDim.x * blockDim.x;
  for (int r = blockIdx.x * blockDim.x + threadIdx.x; r < NROWS; r += stride) {
    int l[AU]; float s[AU];
    const int base = r * AU;
    load_row_i(labels, base, l);
    load_row_f(scores, base, s);
    #pragma unroll
    for (int c = 0; c < AU; c++) {
      if (l[c] == 1) pc[c]++;
      else if (l[c] == -1) {
        nc[c]++;
        atomicAdd(&sh_hist[c * NBINS + score_bin(s[c])], 1);
      }
    }
  }
  #pragma unroll
  for (int c = 0; c < AU; c++) {
    if (pc[c]) atomicAdd(&sh_cnt[c], pc[c]);
    if (nc[c]) atomicAdd(&sh_cnt[AU + c], nc[c]);
  }
  __syncthreads();
  if (threadIdx.x < AU) {
    if (sh_cnt[threadIdx.x]) atomicAdd(&g_pos[threadIdx.x], sh_cnt[threadIdx.x]);
  } else if (threadIdx.x < 2 * AU) {
    int c = threadIdx.x - AU;
    if (sh_cnt[AU + c]) atomicAdd(&g_neg[c], sh_cnt[AU + c]);
  }
  for (int i = threadIdx.x; i < AU * NBINS; i += blockDim.x) {
    int h = sh_hist[i];
    if (h) atomicAdd(&g_hist[i], h);
  }
}

// ---- pass 2: per-column params + boundary bin from histogram CDF (tiny) ----
__global__ void k_params(const int* __restrict__ g_pos, const int* __restrict__ g_neg,
                         const int* __restrict__ g_hist,
                         int* __restrict__ kdrop, int* __restrict__ pflag,
                         float* __restrict__ ratio,
                         int* __restrict__ bbin, int* __restrict__ resid) {
  int j = threadIdx.x;
  if (j >= AU) return;
  float pos = (float)g_pos[j];
  float neg = (float)g_neg[j];
  float half = (pos + neg) * c_balance[j];
  float sample = neg - ceilf(half);
  bool cond = (pos < half) && (sample >= 1.0f);
  int k = cond ? (int)sample : 0;
  kdrop[j] = k;
  pflag[j] = (cond && pos > 0.0f) ? 1 : 0;
  ratio[j] = fminf((pos > 0.0f ? half / fmaxf(pos, 1.0f) : 1.0f), 1.0f);

  int b = -1, r = 0;
  if (k > 0) {
    int cum = 0;
    for (int bin = 0; bin < NBINS; bin++) {
      int h = g_hist[j * NBINS + bin];
      if (cum + h >= k) { b = bin; r = k - cum; break; }
      cum += h;
    }
  }
  bbin[j] = b;
  resid[j] = r;
}

// ---- pass 3: collect (score,row) of negatives landing in the boundary bin ----
__global__ void k_collect(const int* __restrict__ labels, const float* __restrict__ scores,
                          const int* __restrict__ kdrop, const int* __restrict__ bbin,
                          int* __restrict__ tiecnt,
                          float* __restrict__ ties_s, int* __restrict__ ties_r) {
  int Kd[AU], Bb[AU];
  #pragma unroll
  for (int c = 0; c < AU; c++) { Kd[c] = kdrop[c]; Bb[c] = bbin[c]; }
  const int stride = gridDim.x * blockDim.x;
  for (int r = blockIdx.x * blockDim.x + threadIdx.x; r < NROWS; r += stride) {
    int l[AU]; float s[AU];
    const int base = r * AU;
    load_row_i(labels, base, l);
    load_row_f(scores, base, s);
    #pragma unroll
    for (int c = 0; c < AU; c++) {
      if (l[c] == -1 && Kd[c] > 0 && score_bin(s[c]) == Bb[c]) {
        int idx = atomicAdd(&tiecnt[c], 1);
        if (idx < CAP) { ties_s[c * CAP + idx] = s[c]; ties_r[c * CAP + idx] = r; }
      }
    }
  }
}

// ---- pass 4: exact resid-th smallest (score,row) inside the boundary bin ----
__global__ void k_select(const int* __restrict__ kdrop, const int* __restrict__ resid,
                         const int* __restrict__ tiecnt,
                         const float* __restrict__ ties_s, const int* __restrict__ ties_r,
                         float* __restrict__ Tthr, int* __restrict__ Rrow) {
  const int col = blockIdx.x;
  const int k = kdrop[col];
  const int r = resid[col];
  int n = tiecnt[col]; if (n > CAP) n = CAP;
  if (k == 0 || r <= 0) { if (threadIdx.x == 0) { Tthr[col] = -1.0f; Rrow[col] = -1; } return; }
  if (r > n)            { if (threadIdx.x == 0) { Tthr[col] = 2.0f;  Rrow[col] = 0x7fffffff; } return; }
  const float* ts = &ties_s[col * CAP];
  const int*   tr = &ties_r[col * CAP];
  for (int i = threadIdx.x; i < n; i += blockDim.x) {
    float si = ts[i]; int ri = tr[i];
    int cnt = 0;
    for (int q = 0; q < n; q++) {
      float sq = ts[q]; int rq = tr[q];
      cnt += (sq < si) || (sq == si && rq < ri);
    }
    if (cnt == r - 1) { Tthr[col] = si; Rrow[col] = ri; }  // unique writer
  }
}

// ---- pass 5: masked BCE + deterministic partial sums (WMMA wave reduction) ----
__global__ void k_bce(const float* __restrict__ x, const int* __restrict__ labels,
                      const float* __restrict__ scores,
                      const int* __restrict__ kdrop, const int* __restrict__ pflag,
                      const int* __restrict__ bbin, const float* __restrict__ ratio,
                      const float* __restrict__ Tthr, const int* __restrict__ Rrow,
                      float* __restrict__ partial) {
  int Kd[AU], Pf[AU], Bb[AU], Rr[AU]; float Tt[AU], Ra[AU];
  #pragma unroll
  for (int c = 0; c < AU; c++) {
    Kd[c] = kdrop[c]; Pf[c] = pflag[c]; Bb[c] = bbin[c];
    Rr[c] = Rrow[c];  Tt[c] = Tthr[c];  Ra[c] = ratio[c];
  }
  float acc = 0.0f;
  const int stride = gridDim.x * blockDim.x;
  for (int r = blockIdx.x * blockDim.x + threadIdx.x; r < NROWS; r += stride) {
    const int base = r * AU;
    if (r + stride < NROWS) {           // streaming prefetch -> global_prefetch_b8
      __builtin_prefetch(x + (r + stride) * AU, 0, 0);
      __builtin_prefetch(labels + (r + stride) * AU, 0, 0);
      __builtin_prefetch(scores + (r + stride) * AU, 0, 0);
    }
    int l[AU]; float s[AU]; float xv[AU];
    load_row_i(labels, base, l);
    load_row_f(scores, base, s);
    load_row_f(x, base, xv);
    #pragma unroll
    for (int c = 0; c < AU; c++) {
      const int lab = l[c];
      if (lab == 0) continue;
      float m = 1.0f;
      float tgt;
      if (lab < 0) {                    // negative: maybe dropped
        tgt = 0.0f;
        if (Kd[c] > 0) {
          int bin = score_bin(s[c]);
          if (bin < Bb[c]) m = 0.0f;
          else if (bin == Bb[c]) {
            if (s[c] < Tt[c] || (s[c] == Tt[c] && r <= Rr[c])) m = 0.0f;
          }
        }
      } else {                          // positive: maybe rescaled
        tgt = 1.0f;
        if (Pf[c]) m = Ra[c];
      }
      if (m != 0.0f) {
        float xc = xv[c];
        float bce = fmaxf(xc, 0.0f) - xc * tgt + log1pf(__expf(-fabsf(xc)));
        acc += m * bce;
      }
    }
  }

  // ---- wave reduction via V_WMMA_F32_16X16X4_F32 (EXEC all-ones here) ----
  v2f a; a.x = acc;  a.y = 0.0f;        // A(16x4): lane m -> A[m][0], lane 16+m -> A[m][2]
  v2f b; b.x = 1.0f; b.y = 1.0f;        // B(4x16) = ones
  v8f cz = {0.f,0.f,0.f,0.f,0.f,0.f,0.f,0.f};
  v8f d = __builtin_amdgcn_wmma_f32_16x16x4_f32(false, a, false, b, (short)0, cz, false, false);
  // D[m][n] = acc[m] + acc[m+16];  lanes 0-15 hold rows 0..7, lanes 16-31 rows 8..15
  float t = d[0]+d[1]+d[2]+d[3]+d[4]+d[5]+d[6]+d[7];
  t += __shfl_xor(t, 16);               // combine the two half-wave row groups

  __shared__ float sw[8];
  const int lane = threadIdx.x & 31, wav = threadIdx.x >> 5;
  if (lane == 0) sw[wav] = t;
  __syncthreads();
  if (threadIdx.x == 0) {
    float bs = 0.0f;
    #pragma unroll
    for (int i = 0; i < 8; i++) bs += sw[i];
    partial[blockIdx.x] = bs;           // deterministic: fixed block -> fixed slot
  }
}

// ---- pass 6: deterministic final reduce of per-block partials ----
__global__ void k_reduce(const float* __restrict__ partial, float* __restrict__ out) {
  __shared__ float sh[256];
  float s = 0.0f;
  for (int i = threadIdx.x; i < NPART; i += 256) s += partial[i];
  sh[threadIdx.x] = s;
  __syncthreads();
  for (int off = 128; off > 0; off >>= 1) {
    if (threadIdx.x < off) sh[threadIdx.x] += sh[threadIdx.x + off];
    __syncthreads();
  }
  if (threadIdx.x == 0) out[0] = sh[0];
}

extern "C" void kernel_launch(void* const* d_in, const int* in_sizes, int n_in,
                              void* d_out, int out_size, void* d_ws, size_t ws_size,
                              hipStream_t stream) {
  const float* x      = (const float*)d_in[0];
  const int*   labels = (const int*)d_in[1];
  const float* scores = (const float*)d_in[2];
  char* ws = (char*)d_ws;

  int*   pos    = (int*)(ws + OFF_POS);
  int*   neg    = (int*)(ws + OFF_NEG);
  int*   kdrop  = (int*)(ws + OFF_KDROP);
  int*   pflag  = (int*)(ws + OFF_PFLAG);
  int*   bbin   = (int*)(ws + OFF_BBIN);
  int*   resid  = (int*)(ws + OFF_RESID);
  int*   tiecnt = (int*)(ws + OFF_TIECNT);
  int*   Rrow   = (int*)(ws + OFF_RROW);
  float* ratio  = (float*)(ws + OFF_RATIO);
  float* Tthr   = (float*)(ws + OFF_TTHR);
  int*   hist   = (int*)(ws + OFF_HIST);
  float* ties_s = (float*)(ws + OFF_TIES_S);
  int*   ties_r = (int*)(ws + OFF_TIES_R);
  float* part   = (float*)(ws + OFF_PART);

  hipMemsetAsync(ws, 0, WS_ZERO, stream);  // counters + params + histogram

  k_count_hist<<<512, 256, 0, stream>>>(labels, scores, pos, neg, hist);
  k_params    <<<1, 32, 0, stream>>>(pos, neg, hist, kdrop, pflag, ratio, bbin, resid);
  k_collect   <<<512, 256, 0, stream>>>(labels, scores, kdrop, bbin, tiecnt, ties_s, ties_r);
  k_select    <<<AU, 256, 0, stream>>>(kdrop, resid, tiecnt, ties_s, ties_r, Tthr, Rrow);
  k_bce       <<<NPART, 256, 0, stream>>>(x, labels, scores, kdrop, pflag, bbin, ratio,
                                          Tthr, Rrow, part);
  k_reduce    <<<1, 256, 0, stream>>>(part, (float*)d_out);
}